// ScalableFOGNN_58274116272546
// MI455X (gfx1250) — compile-verified
//
#include <hip/hip_runtime.h>
#include <math.h>

#define DEV __device__ __forceinline__

typedef float v2f __attribute__((ext_vector_type(2)));
typedef float v8f __attribute__((ext_vector_type(8)));

constexpr int D  = 64;
constexpr int E  = 1000000;
constexpr int NR = 100000;
constexpr int TILES_E = E / 16;   // 62500
constexpr int TILES_R = NR / 16;  // 6250
constexpr float BN_EPS = 1e-5f;

// ---- workspace layout (float offsets) ----
constexpr size_t WS_SUMS  = 0;        // 5 groups * 128 (sum[64], sumsq[64])
constexpr size_t WS_FOLD  = 1024;     // 5 slots * 4160 (W2' 64x64 + b2' 64)
constexpr size_t FOLD_SLOT = 4160;
constexpr size_t WS_CNT   = 24576;    // NR floats
constexpr size_t WS_AGG   = 131072;   // NR*64
constexpr size_t WS_ZO    = 6553600;  // NR*64  (total ~12.96M floats ~52MB)

DEV v8f vzero8() { v8f z = {0.f,0.f,0.f,0.f,0.f,0.f,0.f,0.f}; return z; }

DEV v8f wmma4(v2f a, v2f b, v8f c) {
  // D(16x16,f32) = A(16x4,f32) x B(4x16,f32) + C
  return __builtin_amdgcn_wmma_f32_16x16x4_f32(false, a, false, b, (short)0, c, false, false);
}

// A fragments for a 16x64 row-major tile: lane holds row (lane&15),
// K = 4k + kb + {0,1} with kb = 2*(lane>>4)  (ISA 16x4 f32 A layout)
DEV void load_a_global(const float* __restrict__ rowp, int kb, v2f* a) {
#pragma unroll
  for (int k = 0; k < 16; ++k) a[k] = *(const v2f*)(rowp + 4*k + kb);
}
DEV void load_a_lds(const float* slot, int kb, int r16, v2f* a) {
#pragma unroll
  for (int k = 0; k < 16; ++k) a[k] = *(const v2f*)(slot + r16*64 + 4*k + kb);
}

// 16x64 @ 64x16 chunk: B fragment from row-major KxN weights (N=64)
DEV v8f gemm64(const v2f* a, const float* __restrict__ W, int kb, int col) {
  v8f acc = vzero8();
#pragma unroll
  for (int k = 0; k < 16; ++k) {
    v2f b;
    b.x = W[(4*k+kb)*64 + col];
    b.y = W[(4*k+kb+1)*64 + col];
    acc = wmma4(a[k], b, acc);
  }
  return acc;
}

// Prefetch next tile's gathered rows: lanes 0-15 touch byte 0, lanes 16-31
// touch byte 128 of each 256B row -> one global_prefetch_b8 covers the tile.
DEV void prefetch_next_edge_tile(const float* __restrict__ left,
                                 const float* __restrict__ efeat,
                                 const float* __restrict__ right,
                                 const int* __restrict__ src,
                                 const int* __restrict__ dst,
                                 int nt, int r16, int hh) {
  if (nt < TILES_E) {
    const int ne0 = nt * 16;
    const int ndi = dst[ne0 + r16];
    const int nsi = src[ne0 + r16];
    __builtin_prefetch(right + (size_t)ndi * D + hh * 32, 0, 0);
    __builtin_prefetch(efeat + (size_t)(ne0 + r16) * D + hh * 32, 0, 0);
    __builtin_prefetch(left + (size_t)nsi * D + hh * 32, 0, 0);
  }
}

// ================= K1: stats of z_ri, z_e, z_lj + edge counts =================
__global__ __launch_bounds__(128) void k1_edge_stats(
    const float* __restrict__ left, const float* __restrict__ efeat,
    const float* __restrict__ right,
    const int* __restrict__ src, const int* __restrict__ dst,
    const float* __restrict__ rW1, const float* __restrict__ rb1,
    const float* __restrict__ eW1, const float* __restrict__ eb1,
    float* __restrict__ sums, float* __restrict__ cnt)
{
  const int lane = threadIdx.x & 31, r16 = lane & 15, hh = lane >> 4, kb = hh * 2;
  const int wave = (blockIdx.x * blockDim.x + threadIdx.x) >> 5;
  const int nw   = (gridDim.x * blockDim.x) >> 5;

  float s[3][4], s2[3][4];
#pragma unroll
  for (int t = 0; t < 3; ++t)
#pragma unroll
    for (int c = 0; c < 4; ++c) { s[t][c] = 0.f; s2[t][c] = 0.f; }

  for (int tile = wave; tile < TILES_E; tile += nw) {
    const int e0 = tile * 16;
    const int di = dst[e0 + r16];
    const int si = src[e0 + r16];
    prefetch_next_edge_tile(left, efeat, right, src, dst, tile + nw, r16, hh);
    const float* rp[3] = { right + (size_t)di * D, efeat + (size_t)(e0 + r16) * D,
                           left + (size_t)si * D };
    const float* W1[3] = { rW1, eW1, rW1 };
    const float* B1[3] = { rb1, eb1, rb1 };
#pragma unroll
    for (int t = 0; t < 3; ++t) {
      v2f a[16]; load_a_global(rp[t], kb, a);
#pragma unroll
      for (int c = 0; c < 4; ++c) {
        const int col = c*16 + r16;
        v8f acc = gemm64(a, W1[t], kb, col);
        const float b = B1[t][col];
        float ls = 0.f, ls2 = 0.f;
#pragma unroll
        for (int r = 0; r < 8; ++r) { float z = acc[r] + b; ls += z; ls2 += z*z; }
        s[t][c] += ls; s2[t][c] += ls2;
      }
    }
    if (lane < 16) unsafeAtomicAdd(&cnt[di], 1.0f);
  }
#pragma unroll
  for (int t = 0; t < 3; ++t)
#pragma unroll
    for (int c = 0; c < 4; ++c) {
      unsafeAtomicAdd(&sums[t*128 + c*16 + r16],      s[t][c]);
      unsafeAtomicAdd(&sums[t*128 + 64 + c*16 + r16], s2[t][c]);
    }
}

// ============== finalize BN stats + fold into second linear ==============
__global__ __launch_bounds__(64) void k_fold(
    const float* __restrict__ sums_g,  // [128] sum, sumsq for this group
    const float* __restrict__ W2, const float* __restrict__ b2,
    float* __restrict__ W2p, float* __restrict__ b2p, float invN)
{
  __shared__ float mean_s[64], rstd_s[64];
  const int n = threadIdx.x;
  float m = sums_g[n] * invN;
  float v = sums_g[64 + n] * invN - m * m;
  mean_s[n] = m;
  rstd_s[n] = rsqrtf(v + BN_EPS);
  __syncthreads();
  float acc = 0.f;
  for (int k = 0; k < 64; ++k) {
    float w = rstd_s[k] * W2[k*64 + n];
    W2p[k*64 + n] = w;
    acc += mean_s[k] * w;
  }
  b2p[n] = b2[n] - acc;
}

// ====== K3: recompute z, folded 2nd linears, f-GEMM, accumulate f stats ======
__global__ __launch_bounds__(64) void k3_fstats(
    const float* __restrict__ left, const float* __restrict__ efeat,
    const float* __restrict__ right,
    const int* __restrict__ src, const int* __restrict__ dst,
    const float* __restrict__ rW1, const float* __restrict__ rb1,
    const float* __restrict__ eW1, const float* __restrict__ eb1,
    const float* __restrict__ W2p0, const float* __restrict__ b2p0,
    const float* __restrict__ W2p1, const float* __restrict__ b2p1,
    const float* __restrict__ W2p2, const float* __restrict__ b2p2,
    const float* __restrict__ fW1, const float* __restrict__ fb1,
    float* __restrict__ sums_f)
{
  extern __shared__ float smem[];
  const int lane = threadIdx.x & 31, r16 = lane & 15, hh = lane >> 4, kb = hh * 2;
  float* base = smem + (threadIdx.x >> 5) * 3072;   // 3 slots of 16x64
  const int wave = (blockIdx.x * blockDim.x + threadIdx.x) >> 5;
  const int nw   = (gridDim.x * blockDim.x) >> 5;

  float s[4] = {0,0,0,0}, s2[4] = {0,0,0,0};

  for (int tile = wave; tile < TILES_E; tile += nw) {
    const int e0 = tile * 16;
    const int di = dst[e0 + r16];
    const int si = src[e0 + r16];
    prefetch_next_edge_tile(left, efeat, right, src, dst, tile + nw, r16, hh);
    const float* rp[3]  = { right + (size_t)di * D, efeat + (size_t)(e0 + r16) * D,
                            left + (size_t)si * D };
    const float* W1[3]  = { rW1, eW1, rW1 };
    const float* B1[3]  = { rb1, eb1, rb1 };
    const float* W2p[3] = { W2p0, W2p1, W2p2 };
    const float* B2p[3] = { b2p0, b2p1, b2p2 };
#pragma unroll
    for (int t = 0; t < 3; ++t) {
      float* slot = base + t * 1024;
      { // z = x @ W1 + b1  -> LDS (row-major 16x64)
        v2f a[16]; load_a_global(rp[t], kb, a);
#pragma unroll
        for (int c = 0; c < 4; ++c) {
          const int col = c*16 + r16;
          v8f acc = gemm64(a, W1[t], kb, col);
          const float b = B1[t][col];
#pragma unroll
          for (int r = 0; r < 8; ++r) slot[(r + 8*hh)*64 + col] = acc[r] + b;
        }
      }
      { // h = relu(z @ W2' + b2')  -> overwrite slot (all A loads hoisted first)
        v2f a[16]; load_a_lds(slot, kb, r16, a);
#pragma unroll
        for (int c = 0; c < 4; ++c) {
          const int col = c*16 + r16;
          v8f acc = gemm64(a, W2p[t], kb, col);
          const float b = B2p[t][col];
#pragma unroll
          for (int r = 0; r < 8; ++r) slot[(r + 8*hh)*64 + col] = fmaxf(acc[r] + b, 0.f);
        }
      }
    }
    // z_f = concat(hi,he,hj) @ fW1 + fb1 ; stats only
#pragma unroll
    for (int c = 0; c < 4; ++c) {
      const int col = c*16 + r16;
      v8f acc = vzero8();
#pragma unroll
      for (int k = 0; k < 48; ++k) {
        const float* slot = base + (k >> 4) * 1024;
        v2f a = *(const v2f*)(slot + r16*64 + 4*(k & 15) + kb);
        const int kabs = 4*k + kb;
        v2f b; b.x = fW1[kabs*64 + col]; b.y = fW1[(kabs+1)*64 + col];
        acc = wmma4(a, b, acc);
      }
      const float bb = fb1[col];
      float ls = 0.f, ls2 = 0.f;
#pragma unroll
      for (int r = 0; r < 8; ++r) { float z = acc[r] + bb; ls += z; ls2 += z*z; }
      s[c] += ls; s2[c] += ls2;
    }
  }
#pragma unroll
  for (int c = 0; c < 4; ++c) {
    unsafeAtomicAdd(&sums_f[c*16 + r16],      s[c]);
    unsafeAtomicAdd(&sums_f[64 + c*16 + r16], s2[c]);
  }
}

// ====== K5: recompute everything, msg = relu(BNf-folded), atomic aggregate ======
__global__ __launch_bounds__(64) void k5_msg(
    const float* __restrict__ left, const float* __restrict__ efeat,
    const float* __restrict__ right,
    const int* __restrict__ src, const int* __restrict__ dst,
    const float* __restrict__ rW1, const float* __restrict__ rb1,
    const float* __restrict__ eW1, const float* __restrict__ eb1,
    const float* __restrict__ W2p0, const float* __restrict__ b2p0,
    const float* __restrict__ W2p1, const float* __restrict__ b2p1,
    const float* __restrict__ W2p2, const float* __restrict__ b2p2,
    const float* __restrict__ fW1, const float* __restrict__ fb1,
    const float* __restrict__ fW2p, const float* __restrict__ fb2p,
    float* __restrict__ agg)
{
  extern __shared__ float smem[];
  const int lane = threadIdx.x & 31, r16 = lane & 15, hh = lane >> 4, kb = hh * 2;
  float* base = smem + (threadIdx.x >> 5) * 4096;   // 3 h slots + 1 zf slot
  const int wave = (blockIdx.x * blockDim.x + threadIdx.x) >> 5;
  const int nw   = (gridDim.x * blockDim.x) >> 5;

  for (int tile = wave; tile < TILES_E; tile += nw) {
    const int e0 = tile * 16;
    const int di = dst[e0 + r16];
    const int si = src[e0 + r16];
    prefetch_next_edge_tile(left, efeat, right, src, dst, tile + nw, r16, hh);
    const float* rp[3]  = { right + (size_t)di * D, efeat + (size_t)(e0 + r16) * D,
                            left + (size_t)si * D };
    const float* W1[3]  = { rW1, eW1, rW1 };
    const float* B1[3]  = { rb1, eb1, rb1 };
    const float* W2p[3] = { W2p0, W2p1, W2p2 };
    const float* B2p[3] = { b2p0, b2p1, b2p2 };
#pragma unroll
    for (int t = 0; t < 3; ++t) {
      float* slot = base + t * 1024;
      {
        v2f a[16]; load_a_global(rp[t], kb, a);
#pragma unroll
        for (int c = 0; c < 4; ++c) {
          const int col = c*16 + r16;
          v8f acc = gemm64(a, W1[t], kb, col);
          const float b = B1[t][col];
#pragma unroll
          for (int r = 0; r < 8; ++r) slot[(r + 8*hh)*64 + col] = acc[r] + b;
        }
      }
      {
        v2f a[16]; load_a_lds(slot, kb, r16, a);
#pragma unroll
        for (int c = 0; c < 4; ++c) {
          const int col = c*16 + r16;
          v8f acc = gemm64(a, W2p[t], kb, col);
          const float b = B2p[t][col];
#pragma unroll
          for (int r = 0; r < 8; ++r) slot[(r + 8*hh)*64 + col] = fmaxf(acc[r] + b, 0.f);
        }
      }
    }
    float* zf = base + 3 * 1024;
#pragma unroll
    for (int c = 0; c < 4; ++c) {
      const int col = c*16 + r16;
      v8f acc = vzero8();
#pragma unroll
      for (int k = 0; k < 48; ++k) {
        const float* slot = base + (k >> 4) * 1024;
        v2f a = *(const v2f*)(slot + r16*64 + 4*(k & 15) + kb);
        const int kabs = 4*k + kb;
        v2f b; b.x = fW1[kabs*64 + col]; b.y = fW1[(kabs+1)*64 + col];
        acc = wmma4(a, b, acc);
      }
      const float bb = fb1[col];
#pragma unroll
      for (int r = 0; r < 8; ++r) zf[(r + 8*hh)*64 + col] = acc[r] + bb;
    }
    int dM[8];
#pragma unroll
    for (int r = 0; r < 8; ++r) dM[r] = dst[e0 + r + 8*hh];
    {
      v2f a[16]; load_a_lds(zf, kb, r16, a);
#pragma unroll
      for (int c = 0; c < 4; ++c) {
        const int col = c*16 + r16;
        v8f acc = gemm64(a, fW2p, kb, col);
        const float b = fb2p[col];
#pragma unroll
        for (int r = 0; r < 8; ++r) {
          float m = fmaxf(acc[r] + b, 0.f);
          unsafeAtomicAdd(&agg[(size_t)dM[r]*64 + col], m);
        }
      }
    }
  }
}

// ====== K6: per-right-node z_o = concat(mean, right) @ oW1 + ob1 ; o-stats ======
__global__ __launch_bounds__(128) void k6_zo(
    const float* __restrict__ right, const float* __restrict__ agg,
    const float* __restrict__ cnt,
    const float* __restrict__ oW1, const float* __restrict__ ob1,
    float* __restrict__ zo, float* __restrict__ sums_o)
{
  const int lane = threadIdx.x & 31, r16 = lane & 15, hh = lane >> 4, kb = hh * 2;
  const int wave = (blockIdx.x * blockDim.x + threadIdx.x) >> 5;
  const int nw   = (gridDim.x * blockDim.x) >> 5;

  float s[4] = {0,0,0,0}, s2[4] = {0,0,0,0};

  for (int tile = wave; tile < TILES_R; tile += nw) {
    const int node = tile * 16 + r16;
    const float invc = 1.0f / fmaxf(cnt[node], 1.0f);
    const float* pa = agg   + (size_t)node * D;
    const float* pr = right + (size_t)node * D;
    v2f a[32];
#pragma unroll
    for (int k = 0; k < 16; ++k) { v2f t = *(const v2f*)(pa + 4*k + kb); a[k] = t * invc; }
#pragma unroll
    for (int k = 0; k < 16; ++k) a[16 + k] = *(const v2f*)(pr + 4*k + kb);
#pragma unroll
    for (int c = 0; c < 4; ++c) {
      const int col = c*16 + r16;
      v8f acc = vzero8();
#pragma unroll
      for (int k = 0; k < 32; ++k) {
        const int kabs = 4*k + kb;  // 0..127
        v2f b; b.x = oW1[kabs*64 + col]; b.y = oW1[(kabs+1)*64 + col];
        acc = wmma4(a[k], b, acc);
      }
      const float bb = ob1[col];
      float ls = 0.f, ls2 = 0.f;
#pragma unroll
      for (int r = 0; r < 8; ++r) {
        float z = acc[r] + bb; ls += z; ls2 += z*z;
        zo[(size_t)(tile*16 + r + 8*hh)*64 + col] = z;
      }
      s[c] += ls; s2[c] += ls2;
    }
  }
#pragma unroll
  for (int c = 0; c < 4; ++c) {
    unsafeAtomicAdd(&sums_o[c*16 + r16],      s[c]);
    unsafeAtomicAdd(&sums_o[64 + c*16 + r16], s2[c]);
  }
}

// ====== K8: out = relu(zo @ oW2' + ob2') ======
__global__ __launch_bounds__(128) void k8_out(
    const float* __restrict__ zo,
    const float* __restrict__ oW2p, const float* __restrict__ ob2p,
    float* __restrict__ out)
{
  const int lane = threadIdx.x & 31, r16 = lane & 15, hh = lane >> 4, kb = hh * 2;
  const int wave = (blockIdx.x * blockDim.x + threadIdx.x) >> 5;
  const int nw   = (gridDim.x * blockDim.x) >> 5;

  for (int tile = wave; tile < TILES_R; tile += nw) {
    v2f a[16]; load_a_global(zo + (size_t)(tile*16 + r16) * D, kb, a);
#pragma unroll
    for (int c = 0; c < 4; ++c) {
      const int col = c*16 + r16;
      v8f acc = gemm64(a, oW2p, kb, col);
      const float b = ob2p[col];
#pragma unroll
      for (int r = 0; r < 8; ++r)
        out[(size_t)(tile*16 + r + 8*hh)*64 + col] = fmaxf(acc[r] + b, 0.f);
    }
  }
}

extern "C" void kernel_launch(void* const* d_in, const int* in_sizes, int n_in,
                              void* d_out, int out_size, void* d_ws, size_t ws_size,
                              hipStream_t stream) {
  const float* left  = (const float*)d_in[0];
  const int*   eidx  = (const int*)d_in[1];
  const float* efeat = (const float*)d_in[2];
  const float* right = (const float*)d_in[3];
  const float* rW1 = (const float*)d_in[4],  *rb1 = (const float*)d_in[5];
  const float* rW2 = (const float*)d_in[6],  *rb2 = (const float*)d_in[7];
  const float* eW1 = (const float*)d_in[8],  *eb1 = (const float*)d_in[9];
  const float* eW2 = (const float*)d_in[10], *eb2 = (const float*)d_in[11];
  const float* fW1 = (const float*)d_in[12], *fb1 = (const float*)d_in[13];
  const float* fW2 = (const float*)d_in[14], *fb2 = (const float*)d_in[15];
  const float* oW1 = (const float*)d_in[16], *ob1 = (const float*)d_in[17];
  const float* oW2 = (const float*)d_in[18], *ob2 = (const float*)d_in[19];
  const int* src = eidx;       // edge_indices[0]
  const int* dst = eidx + E;   // edge_indices[1]

  float* ws   = (float*)d_ws;
  float* sums = ws + WS_SUMS;
  float* cnt  = ws + WS_CNT;
  float* agg  = ws + WS_AGG;
  float* zo   = ws + WS_ZO;
  float* out  = (float*)d_out;
  auto foldW = [&](int s) { return ws + WS_FOLD + (size_t)s * FOLD_SLOT; };
  auto foldB = [&](int s) { return ws + WS_FOLD + (size_t)s * FOLD_SLOT + 4096; };

  hipMemsetAsync(sums, 0, 1024 * sizeof(float), stream);
  hipMemsetAsync(cnt,  0, (size_t)NR * sizeof(float), stream);
  hipMemsetAsync(agg,  0, (size_t)NR * D * sizeof(float), stream);

  // K1: first-linear stats for hi/he/hj + counts
  k1_edge_stats<<<1024, 128, 0, stream>>>(left, efeat, right, src, dst,
                                          rW1, rb1, eW1, eb1, sums, cnt);
  // fold ri / e / lj second linears
  k_fold<<<1, 64, 0, stream>>>(sums + 0,   rW2, rb2, foldW(0), foldB(0), 1.0f / E);
  k_fold<<<1, 64, 0, stream>>>(sums + 128, eW2, eb2, foldW(1), foldB(1), 1.0f / E);
  k_fold<<<1, 64, 0, stream>>>(sums + 256, rW2, rb2, foldW(2), foldB(2), 1.0f / E);
  // K3: f-stage stats
  k3_fstats<<<4096, 64, 2 * 3072 * sizeof(float), stream>>>(
      left, efeat, right, src, dst, rW1, rb1, eW1, eb1,
      foldW(0), foldB(0), foldW(1), foldB(1), foldW(2), foldB(2),
      fW1, fb1, sums + 384);
  k_fold<<<1, 64, 0, stream>>>(sums + 384, fW2, fb2, foldW(3), foldB(3), 1.0f / E);
  // K5: messages + mean-aggregation numerator
  k5_msg<<<4096, 64, 2 * 4096 * sizeof(float), stream>>>(
      left, efeat, right, src, dst, rW1, rb1, eW1, eb1,
      foldW(0), foldB(0), foldW(1), foldB(1), foldW(2), foldB(2),
      fW1, fb1, foldW(3), foldB(3), agg);
  // K6: output-module first linear + stats
  k6_zo<<<256, 128, 0, stream>>>(right, agg, cnt, oW1, ob1, zo, sums + 512);
  k_fold<<<1, 64, 0, stream>>>(sums + 512, oW2, ob2, foldW(4), foldB(4), 1.0f / NR);
  // K8: final output
  k8_out<<<256, 128, 0, stream>>>(zo, foldW(4), foldB(4), out);
  (void)in_sizes; (void)n_in; (void)out_size; (void)ws_size;
}